// SiameseLossLayer_32847909880368
// MI455X (gfx1250) — compile-verified
//
#include <hip/hip_runtime.h>

// SiameseLossLayer: sum(relu(inputs[I] - inputs[J])) over margin pairs.
// inputs: [7680, 2048] f32.  Pairs: base = b*d (b<64, d<24), levels i<j in [0,5).
// (i==j pairs contribute exactly 0 and are skipped.)

typedef __attribute__((ext_vector_type(2))) float v2f;
typedef __attribute__((ext_vector_type(8))) float v8f;

#define FEAT    2048
#define NLEV    5
#define NPAIRS  10        // i<j pairs among 5 levels
#define NGROUPS 1536      // 64 * 24 base groups
#define BLOCK   320       // 10 waves (wave32), one wave per pair

// ii = {0,0,0,0,1,1,1,2,2,3} packed 2 bits each
#define II_PACK 0x000E9500u
// jj = {1,2,3,4,2,3,4,3,4,4} packed 3 bits each
#define JJ_PACK 0x2471A8D1u

__global__ void siamese_zero_kernel(float* out) {
    if (threadIdx.x == 0) out[0] = 0.0f;
}

__global__ __launch_bounds__(BLOCK)
void siamese_hinge_sum_kernel(const float* __restrict__ in, float* __restrict__ out) {
    __shared__ float rows[NLEV * FEAT];   // 40 KB: 5 contiguous rows of this group
    __shared__ float wave_part[NPAIRS];

    const int g    = blockIdx.x;
    const int b    = g / 24;
    const int d    = g - b * 24;
    const int base = b * d;               // bug-faithful reference indexing

    // ---- Stage rows base..base+4 (contiguous 40 KB) into LDS, float4-coalesced.
    const float4* __restrict__ src = (const float4*)(in + (size_t)base * FEAT);
    float4* dst = (float4*)rows;
    const int t = threadIdx.x;
    #pragma unroll
    for (int k = 0; k < (NLEV * FEAT / 4) / BLOCK; ++k)   // 8 iters
        dst[t + k * BLOCK] = src[t + k * BLOCK];
    __syncthreads();

    // ---- One wave per (i, j) pair.
    const int wave = t >> 5;
    const int lane = t & 31;
    const int i = (II_PACK >> (2 * wave)) & 3;
    const int j = (JJ_PACK >> (3 * wave)) & 7;

    const float* rowI = rows + i * FEAT;
    const float* rowJ = rows + j * FEAT;

    // f32 WMMA as an exact reduction engine: D = A x ones(4x16) + C.
    v8f acc = {};
    v2f onesv; onesv.x = 1.0f; onesv.y = 1.0f;

    #pragma unroll
    for (int it = 0; it < FEAT / 128; ++it) {             // 16 iters, uniform for all lanes
        const int f = it * 128 + lane * 4;
        float4 a = *(const float4*)(rowI + f);            // ds_load_b128
        float4 c = *(const float4*)(rowJ + f);            // ds_load_b128
        float r0 = fmaxf(a.x - c.x, 0.0f);
        float r1 = fmaxf(a.y - c.y, 0.0f);
        float r2 = fmaxf(a.z - c.z, 0.0f);
        float r3 = fmaxf(a.w - c.w, 0.0f);
        v2f av; av.x = r0 + r1; av.y = r2 + r3;
        // v_wmma_f32_16x16x4_f32: each A element lands in every N column of D,
        // so acc over-counts by exactly 16x (corrected below).
        acc = __builtin_amdgcn_wmma_f32_16x16x4_f32(
            /*neg_a=*/false, av, /*neg_b=*/false, onesv,
            /*c_mod=*/(short)0, acc, /*reuse_a=*/false, /*reuse_b=*/false);
    }

    // ---- Per-lane fold of the C tile, then wave32 shuffle reduction.
    float s = acc[0] + acc[1] + acc[2] + acc[3] + acc[4] + acc[5] + acc[6] + acc[7];
    #pragma unroll
    for (int off = 16; off > 0; off >>= 1)
        s += __shfl_down(s, off, 32);

    if (lane == 0) wave_part[wave] = s * 0.0625f;         // undo 16x column replication
    __syncthreads();

    // ---- One atomic per block (1536 total), cheap at L2.
    if (t == 0) {
        float bs = 0.0f;
        #pragma unroll
        for (int w = 0; w < NPAIRS; ++w) bs += wave_part[w];
        atomicAdd(out, bs);
    }
}

extern "C" void kernel_launch(void* const* d_in, const int* in_sizes, int n_in,
                              void* d_out, int out_size, void* d_ws, size_t ws_size,
                              hipStream_t stream) {
    (void)in_sizes; (void)n_in; (void)out_size; (void)d_ws; (void)ws_size;
    const float* in = (const float*)d_in[0];
    float* out = (float*)d_out;

    siamese_zero_kernel<<<1, 32, 0, stream>>>(out);
    siamese_hinge_sum_kernel<<<NGROUPS, BLOCK, 0, stream>>>(in, out);
}